// AtomFeature_85031762526727
// MI455X (gfx1250) — compile-verified
//
#include <hip/hip_runtime.h>
#include <math.h>

typedef __attribute__((ext_vector_type(2))) float v2f;
typedef __attribute__((ext_vector_type(4))) float v4f;
typedef __attribute__((ext_vector_type(8))) float v8f;

#define NUM_MAIN   12
#define EMB_DIM    32
#define KNN        32
#define BIGV       1000000.0f          // reference BIG (output domain)
#define MASKKEY    1.0e12f             // key for masked pairs (squared domain)
#define TAKENKEY   4.0e12f             // key for already-extracted slots
#define INITKEY    2.0e12f             // scan initializer (> any valid key)
#define NATOMS     4608
#define ROWS       16                  // query rows per workgroup (one WMMA M-tile)
#define SEGS       2                   // column segments staged in LDS
#define SEGCOLS    (NATOMS / SEGS)     // 2304
#define SEGTILES   (SEGCOLS / 16)      // 144 WMMA column tiles per segment
#define STRIPE     (SEGCOLS / 16)      // 144 elems per lane strip
#define SPAD       148                 // padded strip (16B-aligned, bank-spread)
#define RSTRIDE    (16 * SPAD)         // per-row DWORD stride (2368)
#define CAND       (KNN * SEGS)        // 64 merge candidates per row

struct F3 { float x, y, z; };          // 12-byte coord record -> b96/b64 loads

// ---------------------------------------------------------------------------
// Kernel 1: embedding lookup + graph_norm.  One block per (batch, channel).
// ---------------------------------------------------------------------------
__global__ void embed_norm_kernel(const float* __restrict__ mask,   // (B,N)
                                  const float* __restrict__ table,  // (12,32)
                                  const float* __restrict__ scale,  // (32)
                                  const float* __restrict__ shift,  // (32)
                                  float* __restrict__ out,          // (B,N,32)
                                  int N) {
  const int b   = blockIdx.x;
  const int d   = blockIdx.y;
  const int tid = threadIdx.x;

  float s1 = 0.0f, s2 = 0.0f, cnt = 0.0f;
  for (int n = tid; n < N; n += blockDim.x) {
    float mk = mask[b * N + n];
    float v  = table[(n % NUM_MAIN) * EMB_DIM + d] * mk;
    s1 += v; s2 += v * v; cnt += mk;
  }
  #pragma unroll
  for (int off = 16; off > 0; off >>= 1) {
    s1  += __shfl_xor(s1,  off, 32);
    s2  += __shfl_xor(s2,  off, 32);
    cnt += __shfl_xor(cnt, off, 32);
  }
  __shared__ float rs1[8], rs2[8], rc[8];
  const int wave = tid >> 5;
  if ((tid & 31) == 0) { rs1[wave] = s1; rs2[wave] = s2; rc[wave] = cnt; }
  __syncthreads();
  float S1 = 0.0f, S2 = 0.0f, C = 0.0f;
  const int nwaves = blockDim.x >> 5;
  for (int w = 0; w < nwaves; ++w) { S1 += rs1[w]; S2 += rs2[w]; C += rc[w]; }

  const float cc   = fmaxf(C, 1.0f);
  const float mean = S1 / cc;
  const float var  = (S2 - 2.0f * mean * S1 + (float)N * mean * mean) / cc;
  const float istd = 1.0f / sqrtf(var + 1e-5f);
  const float sc   = scale[d];
  const float sh   = shift[d];

  for (int n = tid; n < N; n += blockDim.x) {
    float mk = mask[b * N + n];
    float f  = table[(n % NUM_MAIN) * EMB_DIM + d] * mk;
    out[(size_t)(b * N + n) * EMB_DIM + d] = ((f - mean) * istd * sc + sh) * mk;
  }
}

// ---------------------------------------------------------------------------
// Kernel 2: kNN graph via WMMA squared-distance tiles + LDS top-K selection.
//   grid = (N/16, B), block = 256 threads (8 waves).
//   d2[m][n] = WMMA( A_m=(x,y,z,1) , B_n=(-2x,-2y,-2z,|x_n|^2) ) + |x_m|^2
//   Column masking folded into B (masked col -> B=(0,0,0,MASKKEY)).
//   Key slab stored STRIP-MAJOR per row: column c -> slot (c&15)*SPAD+(c>>4),
//   so each lane's selection strip is contiguous and scans with b128 loads.
//   Selection on squared keys; sqrt only for the 32 winners per row.
// ---------------------------------------------------------------------------
__global__ __launch_bounds__(256) void knn_wmma_kernel(
    const float* __restrict__ coords,   // (B,N,3)
    const float* __restrict__ mask,     // (B,N)
    float*       __restrict__ outDist,  // (B,N,K)
    long long*   __restrict__ outIdx) { // (B,N,K)
  extern __shared__ char smem_raw[];
  float* sKey   = (float*)smem_raw;                 // ROWS * RSTRIDE
  float* sCandD = sKey + ROWS * RSTRIDE;            // ROWS * CAND
  int*   sCandI = (int*)(sCandD + ROWS * CAND);     // ROWS * CAND
  float* sQ     = (float*)(sCandI + ROWS * CAND);   // ROWS * 4 : x,y,z,|x|^2
  float* sQm    = sQ + ROWS * 4;                    // ROWS

  const F3* c3 = reinterpret_cast<const F3*>(coords);

  const int b       = blockIdx.y;
  const int rowBase = blockIdx.x * ROWS;
  const int tid     = threadIdx.x;
  const int lane    = tid & 31;
  const int wave    = tid >> 5;
  const int hi      = lane >> 4;   // 0: lanes 0-15, 1: lanes 16-31
  const int lm      = lane & 15;

  if (tid < ROWS) {
    int q = rowBase + tid;
    F3 p = c3[(size_t)b * NATOMS + q];
    sQ[tid * 4 + 0] = p.x;
    sQ[tid * 4 + 1] = p.y;
    sQ[tid * 4 + 2] = p.z;
    sQ[tid * 4 + 3] = p.x * p.x + p.y * p.y + p.z * p.z;
    sQm[tid] = mask[(size_t)b * NATOMS + q];
  }
  __syncthreads();

  // A operand (16x4 f32): lane m<16 -> {x_m, y_m}; lane 16+m -> {z_m, 1.0}
  v2f A;
  A.x = hi ? sQ[lm * 4 + 2] : sQ[lm * 4 + 0];
  A.y = hi ? 1.0f           : sQ[lm * 4 + 1];

  float rq[8];
  #pragma unroll
  for (int v = 0; v < 8; ++v) rq[v] = sQ[(v + hi * 8) * 4 + 3];

  for (int seg = 0; seg < SEGS; ++seg) {
    const int segBase = seg * SEGCOLS;

    // ---- phase 1: fill strip-major squared-distance keys via WMMA ----
    for (int t = wave; t < SEGTILES; t += 8) {
      const int j = segBase + t * 16 + lm;       // candidate column
      F3 p = c3[(size_t)b * NATOMS + j];
      bool mok = (mask[(size_t)b * NATOMS + j] != 0.0f);
      float n2 = p.x * p.x + p.y * p.y + p.z * p.z;
      v2f Bv;                                     // branchless (cndmask) setup
      Bv.x = mok ? (hi ? -2.0f * p.z : -2.0f * p.x) : 0.0f;
      Bv.y = mok ? (hi ? n2          : -2.0f * p.y) : (hi ? MASKKEY : 0.0f);

      v8f acc = {};
      acc = __builtin_amdgcn_wmma_f32_16x16x4_f32(
          /*neg_a=*/false, A, /*neg_b=*/false, Bv,
          /*c_mod=*/(short)0, acc, /*reuse_a=*/false, /*reuse_b=*/false);

      // column c = t*16+lm -> strip lm, position t (strip-major layout)
      float* dst = sKey + hi * 8 * RSTRIDE + lm * SPAD + t;
      #pragma unroll
      for (int v = 0; v < 8; ++v) {
        float key = fmaxf(acc[v] + rq[v], 0.0f);   // -2<xi,xj>+|xj|^2+|xi|^2
        dst[v * RSTRIDE] = key;
      }
    }
    __syncthreads();

    // ---- phase 2: per-row top-32, half-wave per row, b128 strip scan ----
    // lanes 0-15 own row `wave`, lanes 16-31 own row `wave+8`.
    {
      const int r = wave + hi * 8;
      float* row   = sKey + r * RSTRIDE;
      const float* strip = row + lm * SPAD;        // contiguous 144 floats
      for (int k = 0; k < KNN; ++k) {
        float bd   = INITKEY;
        int   bpos = 0;
        for (int i = 0; i < STRIPE / 4; ++i) {     // 36 x ds_load_b128
          v4f q4 = *(const v4f*)(strip + (i << 2));
          #pragma unroll
          for (int e = 0; e < 4; ++e) {
            float dv  = q4[e];
            int   pos = (i << 2) + e;
            // strict `<`: ascending pos keeps lowest index on ties
            if (dv < bd) { bd = dv; bpos = pos; }
          }
        }
        int bi = segBase + (bpos << 4) + lm;       // c = pos*16 + strip
        #pragma unroll
        for (int off = 8; off > 0; off >>= 1) {    // stays within half-wave
          float od = __shfl_xor(bd, off, 32);
          int   oi = __shfl_xor(bi, off, 32);
          if (od < bd || (od == bd && oi < bi)) { bd = od; bi = oi; }
        }
        if (lm == 0) {
          int cl = bi - segBase;                   // local column
          row[(cl & 15) * SPAD + (cl >> 4)] = TAKENKEY;   // mark taken
          sCandD[r * CAND + seg * KNN + k] = bd;
          sCandI[r * CAND + seg * KNN + k] = bi;
        }
        __builtin_amdgcn_wave_barrier();           // order LDS RMW rounds
      }
    }
    __syncthreads();
  }

  // ---- phase 3: merge 64 candidates -> 32 smallest, ascending; emit ----
  for (int rr = 0; rr < 2; ++rr) {
    const int r  = wave + rr * 8;
    const int q  = rowBase + r;
    const float qm = sQm[r];
    float* cd = sCandD + r * CAND;
    int*   ci = sCandI + r * CAND;
    for (int k = 0; k < KNN; ++k) {
      float d0 = cd[lane];      int i0 = ci[lane];
      float d1 = cd[lane + 32]; int i1 = ci[lane + 32];
      float bd; int bi; int bslot;
      if (d1 < d0 || (d1 == d0 && i1 < i0)) { bd = d1; bi = i1; bslot = lane + 32; }
      else                                  { bd = d0; bi = i0; bslot = lane; }
      #pragma unroll
      for (int off = 16; off > 0; off >>= 1) {
        float od = __shfl_xor(bd, off, 32);
        int   oi = __shfl_xor(bi, off, 32);
        int   os = __shfl_xor(bslot, off, 32);
        if (od < bd || (od == bd && oi < bi)) { bd = od; bi = oi; bslot = os; }
      }
      if (lane == 0) {
        cd[bslot] = TAKENKEY;
        bool pad    = (qm == 0.0f);
        bool masked = (bd >= 0.5f * MASKKEY);      // masked pair -> BIG in ref
        float dist  = masked ? BIGV : sqrtf(bd + 1e-6f);
        outDist[((size_t)b * NATOMS + q) * KNN + k] = pad ? BIGV : dist;
        outIdx [((size_t)b * NATOMS + q) * KNN + k] = pad ? -1ll : (long long)bi;
      }
      __builtin_amdgcn_wave_barrier();
    }
  }
}

// ---------------------------------------------------------------------------
extern "C" void kernel_launch(void* const* d_in, const int* in_sizes, int n_in,
                              void* d_out, int out_size, void* d_ws, size_t ws_size,
                              hipStream_t stream) {
  const float* coords = (const float*)d_in[0];   // (B,N,3)
  const float* mask   = (const float*)d_in[1];   // (B,N)
  const float* table  = (const float*)d_in[2];   // (12,32)
  const float* scale  = (const float*)d_in[3];   // (32)
  const float* shift  = (const float*)d_in[4];   // (32)

  const int B = in_sizes[1] / NATOMS;

  float*     outEmb  = (float*)d_out;                                  // B*N*32
  float*     outDist = outEmb + (size_t)B * NATOMS * EMB_DIM;          // B*N*32
  long long* outIdx  = (long long*)(outDist + (size_t)B * NATOMS * KNN);

  embed_norm_kernel<<<dim3(B, EMB_DIM), 256, 0, stream>>>(
      mask, table, scale, shift, outEmb, NATOMS);

  const size_t shmem = (size_t)ROWS * RSTRIDE * sizeof(float)   // key slab (148 KB)
                     + (size_t)ROWS * CAND * sizeof(float)      // cand keys
                     + (size_t)ROWS * CAND * sizeof(int)        // cand idx
                     + (size_t)ROWS * 4 * sizeof(float)         // query xyz+norm
                     + (size_t)ROWS * sizeof(float);            // query mask
  knn_wmma_kernel<<<dim3(NATOMS / ROWS, B), 256, shmem, stream>>>(
      coords, mask, outDist, outIdx);
}